// PointPillarMultiViewsProjector_44547400794345
// MI455X (gfx1250) — compile-verified
//
#include <hip/hip_runtime.h>
#include <hip/hip_bf16.h>
#include <math.h>

// Reference constants:
// VOXEL=(0.2,0.2)  PC0=(-51.2,-51.2)  CENTER=(0,0)
// CYL0=(0,-pi)     CYL_VS=(0.3, 2*pi/500)
// pview: (B=4, T=4, P=512, R=256, C=64)   out: (B=4, C=64, H=512, W=512)
#define PI_F 3.14159265358979323846f

typedef unsigned int u32x4 __attribute__((ext_vector_type(4)));
typedef unsigned int u32x8 __attribute__((ext_vector_type(8)));

__device__ __forceinline__ unsigned lds_off_of(const void* p) {
  // generic pointer to LDS = {aperture_hi32, lds_offset}; low 32 bits = DS byte offset
  return (unsigned)(size_t)p;
}

// ---- Tensor DMA descriptor builders (CDNA5 ISA) ----
// Group 0: [1:0]=count=1, [63:32]=lds_addr, [120:64]=global_addr, [127:126]=type=2
__device__ __forceinline__ u32x4 tdm_g0(unsigned lds_addr, unsigned long long gaddr) {
  u32x4 g;
  g.x = 1u;                                    // count=1, user descriptor
  g.y = lds_addr;                              // LDS byte address
  g.z = (unsigned)gaddr;                       // global_addr[31:0]
  g.w = (unsigned)((gaddr >> 32) & 0x01FFFFFFu) | (2u << 30);  // addr[56:32] | type=2
  return g;
}
// Group 1: data_size=4B (code 2); 1-row tile: tensor_dim0=tile_dim0=n0,
// tensor_dim1=tile_dim1=1, tile_dim2=0, strides=n0. No padding/iteration/multicast.
__device__ __forceinline__ u32x8 tdm_g1(unsigned n0 /* elements, <=65535 */) {
  u32x8 g;
  g.s0 = 0x00020000u;                          // workgroup_mask=0, data_size=2 (4B)
  g.s1 = (n0 & 0xFFFFu) << 16;                 // abar_addr=0 | tensor_dim0[15:0]
  g.s2 = (n0 >> 16) | (1u << 16);              // tensor_dim0[31:16] | tensor_dim1[15:0]=1
  g.s3 = (n0 & 0xFFFFu) << 16;                 // tensor_dim1[31:16]=0 | tile_dim0=n0
  g.s4 = 1u;                                   // tile_dim1=1 | tile_dim2=0
  g.s5 = n0;                                   // tensor_dim0_stride[31:0]
  g.s6 = (n0 & 0xFFFFu) << 16;                 // stride0[47:32]=0 | stride1[15:0]
  g.s7 = (n0 >> 16);                           // stride1[47:16]
  return g;
}

// ---------------- Kernel 1: base copy spatial_features -> out via TDM ----------------
// One wave per workgroup; each grid-stride iteration moves one 64KB tile:
//   tensor_load_to_lds (global->LDS) ; s_wait_tensorcnt 0 ; tensor_store_from_lds
#define TDM_TILE_ELEMS 16384u /* 64KB of f32 */

__global__ __launch_bounds__(32) void copy_tdm_kernel(const float* __restrict__ src,
                                                      float* __restrict__ dst,
                                                      unsigned ntiles) {
  __shared__ __align__(128) float tdbuf[TDM_TILE_ELEMS];
  const unsigned lds = lds_off_of(tdbuf);
  const u32x8 g1 = tdm_g1(TDM_TILE_ELEMS);

  for (unsigned t = blockIdx.x; t < ntiles; t += gridDim.x) {
    const unsigned long long off = (unsigned long long)t * (TDM_TILE_ELEMS * 4ull);
    const u32x4 g0l = tdm_g0(lds, (unsigned long long)(size_t)src + off);
    const u32x4 g0s = tdm_g0(lds, (unsigned long long)(size_t)dst + off);
    __builtin_amdgcn_s_wait_tensorcnt(0);   // prior iteration's store drained (buffer reuse)
    asm volatile("tensor_load_to_lds %0, %1" :: "s"(g0l), "s"(g1) : "memory");
    __builtin_amdgcn_s_wait_tensorcnt(0);   // tile resident in LDS
    asm volatile("tensor_store_from_lds %0, %1" :: "s"(g0s), "s"(g1) : "memory");
  }
  // implicit S_WAIT_IDLE at S_ENDPGM drains the last store
}

// tail (out_size % TDM_TILE_ELEMS) — plain grid-stride copy
__global__ __launch_bounds__(256) void copy_tail_kernel(const float* __restrict__ src,
                                                        float* __restrict__ dst,
                                                        size_t start, size_t n) {
  for (size_t i = start + (size_t)blockIdx.x * 256u + threadIdx.x; i < n;
       i += (size_t)gridDim.x * 256u)
    dst[i] = src[i];
}

// ---------------- Kernel 2: cylindrical trilinear gather + atomic scatter ------------
// one wave (32 lanes) per point; each lane owns channels {2*lane, 2*lane+1}
__global__ __launch_bounds__(256) void project_kernel(const int* __restrict__ coords,
                                                      const float* __restrict__ pview,
                                                      float* __restrict__ out,
                                                      int npts) {
  const int lane = (int)(threadIdx.x & 31u);
  int n = (int)(blockIdx.x * 8u + (threadIdx.x >> 5));
  n = __builtin_amdgcn_readfirstlane(n);       // wave-uniform -> scalar addressing
  if (n >= npts) return;

  const int4 vc = ((const int4*)coords)[n];    // [b, t, y, x]
  const int bi = vc.x, tz = vc.y, yc = vc.z, xc = vc.w;

  const float fy = (float)yc * 0.2f - 51.2f;
  const float fx = (float)xc * 0.2f - 51.2f;
  const float r  = sqrtf(fx * fx + fy * fy);
  float psi;
  if (fx == 0.0f && fy >= 0.0f) psi = 0.5f * PI_F;
  else                          psi = atan2f(fy, fx);

  const float r_idx = r * (1.0f / 0.3f);
  const float p_idx = (psi + PI_F) * (500.0f / (2.0f * PI_F));
  const float t_idx = (float)tz;

  const int r0 = (int)floorf(r_idx);
  const int p0 = (int)floorf(p_idx);
  const int t0 = (int)floorf(t_idx);
  const int r1 = min(r0 + 1, 255);
  const int p1 = min(p0 + 1, 511);
  const int t1 = min(t0 + 1, 3);

  const float fr = r_idx - (float)r0;
  const float fp = p_idx - (float)p0;
  const float ft = t_idx - (float)t0;

  const float wr[2] = {1.0f - fr, fr};
  const float wp[2] = {1.0f - fp, fp};
  const float wt[2] = {1.0f - ft, ft};
  const int   ri[2] = {r0, r1};
  const int   pj[2] = {p0, p1};
  const int   ti[2] = {t0, t1};

  float acc0 = 0.0f, acc1 = 0.0f;
#pragma unroll
  for (int jt = 0; jt < 2; ++jt)
#pragma unroll
    for (int jp = 0; jp < 2; ++jp)
#pragma unroll
      for (int jr = 0; jr < 2; ++jr) {
        const float w = wr[jr] * wp[jp] * wt[jt];
        // pview[bi, t, p, r, :]; base is wave-uniform (SGPR), lane offset = 2*lane floats
        const size_t base =
            ((((size_t)bi * 4 + (size_t)ti[jt]) * 512 + (size_t)pj[jp]) * 256 +
             (size_t)ri[jr]) * 64;
        const float2 v = *(const float2*)(pview + base + 2 * lane);  // coalesced 256B
        acc0 = fmaf(w, v.x, acc0);
        acc1 = fmaf(w, v.y, acc1);
      }

  // scatter-add: out[bi, ch, yc, tz], ch = 2*lane, 2*lane+1 ; duplicates -> HW f32 atomics
  const size_t o0 = (((size_t)bi * 64 + (size_t)(2 * lane)) * 512 + (size_t)yc) * 512 +
                    (size_t)tz;
  const unsigned b0 = (unsigned)(o0 << 2);           // byte offsets (<256MB, fits i32)
  const unsigned b1 = (unsigned)((o0 + 512u * 512u) << 2);
  asm volatile("global_atomic_add_f32 %0, %1, %2 scope:SCOPE_DEV"
               :: "v"(b0), "v"(acc0), "s"(out) : "memory");
  asm volatile("global_atomic_add_f32 %0, %1, %2 scope:SCOPE_DEV"
               :: "v"(b1), "v"(acc1), "s"(out) : "memory");
}

// ---------------- launch ----------------
extern "C" void kernel_launch(void* const* d_in, const int* in_sizes, int n_in,
                              void* d_out, int out_size, void* d_ws, size_t ws_size,
                              hipStream_t stream) {
  const int*   coords = (const int*)d_in[0];    // (N,4) int32
  const float* pview  = (const float*)d_in[1];  // (4,4,512,256,64) f32
  const float* spat   = (const float*)d_in[2];  // (4,64,512,512)  f32
  float* out = (float*)d_out;

  const int    npts = in_sizes[0] / 4;
  const size_t nflt = (size_t)out_size;

  // 1) out = spatial_features  via Tensor Data Mover, 64KB tiles (4096 tiles for 256MB)
  const unsigned ntiles = (unsigned)(nflt / TDM_TILE_ELEMS);
  if (ntiles > 0) {
    const unsigned blocks = ntiles < 4096u ? ntiles : 4096u;
    copy_tdm_kernel<<<blocks, 32, 0, stream>>>(spat, out, ntiles);
  }
  const size_t tail_start = (size_t)ntiles * TDM_TILE_ELEMS;
  if (tail_start < nflt) {
    copy_tail_kernel<<<256, 256, 0, stream>>>(spat, out, tail_start, nflt);
  }

  // 2) gather + trilinear blend + atomic scatter-add (1 wave per point)
  const int blocks = (npts + 7) / 8;
  project_kernel<<<blocks, 256, 0, stream>>>(coords, pview, out, npts);
}